// GCN_46213848105685
// MI455X (gfx1250) — compile-verified
//
#include <hip/hip_runtime.h>
#include <math.h>

#define NN 50000
#define NE 800000

typedef float v2f __attribute__((ext_vector_type(2)));
typedef float v8f __attribute__((ext_vector_type(8)));

// Guaranteed CDNA5 hardware float atomic (L2 atomic units, device scope).
__device__ __forceinline__ void atomic_add_f32(float* p, float v) {
    asm volatile("global_atomic_add_f32 %0, %1, off scope:SCOPE_DEV"
                 :
                 : "v"(p), "v"(v)
                 : "memory");
}

// ---------------- degree / normalization ----------------
__global__ void k_deg_init(float* deg) {
    int i = blockIdx.x * blockDim.x + threadIdx.x;
    if (i < NN) deg[i] = 1.0f;   // self-loop contributes 1
}

__global__ void k_deg_accum(const int* __restrict__ dst, float* __restrict__ deg) {
    int e = blockIdx.x * blockDim.x + threadIdx.x;
    if (e < NE) atomic_add_f32(&deg[dst[e]], 1.0f);
}

__global__ void k_dinv(float* deg) {
    int i = blockIdx.x * blockDim.x + threadIdx.x;
    if (i < NN) deg[i] = rsqrtf(deg[i]);     // deg >= 1 always
}

__global__ void k_norm(const int* __restrict__ src, const int* __restrict__ dst,
                       const float* __restrict__ dinv, float* __restrict__ norm) {
    int e = blockIdx.x * blockDim.x + threadIdx.x;
    if (e < NE) norm[e] = dinv[src[e]] * dinv[dst[e]];
}

// ---------------- fp32 WMMA GEMM + fused self-loop/bias epilogue ----------------
// H[N x OUT] = X[N x IN] * W[IN x OUT]
// agg[N x OUT] = H * dinv[row]^2 + bias[col]   (scatter kernel then adds edge terms)
// One wave per 16-row tile; the wave computes all OUT columns so X streams once.
template <int IN, int OUT>
__global__ __launch_bounds__(32) void k_gemm_wmma(const float* __restrict__ X,
                                                  const float* __restrict__ W,
                                                  const float* __restrict__ bias,
                                                  const float* __restrict__ dinv,
                                                  float* __restrict__ H,
                                                  float* __restrict__ agg) {
    constexpr int NT = OUT / 16;
    const int mBase = blockIdx.x * 16;
    const int lane  = threadIdx.x;      // 0..31
    const int hf    = lane >> 4;        // lane half: selects K=0,1 vs K=2,3
    const int lr    = lane & 15;

    v8f acc[NT] = {};
    const float* xrow = X + (long long)(mBase + lr) * IN;

    for (int k = 0; k < IN; k += 4) {
        // A 16x4 f32: v0 = {K0 | K2}, v1 = {K1 | K3} across lane halves
        v2f a;
        const float2 af = *(const float2*)(xrow + k + 2 * hf);
        a.x = af.x;
        a.y = af.y;
#pragma unroll
        for (int t = 0; t < NT; ++t) {
            // B 4x16 f32: row K striped across lanes; v0 = {K0 | K2}, v1 = {K1 | K3}
            const int col = t * 16 + lr;
            v2f b;
            b.x = W[(k + 2 * hf) * OUT + col];
            b.y = W[(k + 2 * hf + 1) * OUT + col];
            acc[t] = __builtin_amdgcn_wmma_f32_16x16x4_f32(
                false, a, false, b, (short)0, acc[t], false, false);
        }
    }

    // C/D layout: VGPR r holds M=r (lanes 0-15) and M=r+8 (lanes 16-31), N=lane%16
#pragma unroll
    for (int r = 0; r < 8; ++r) {
        const int row = mBase + r + 8 * hf;
        const float di = dinv[row];
        const float d2 = di * di;
#pragma unroll
        for (int t = 0; t < NT; ++t) {
            const int col = t * 16 + lr;
            const float v = acc[t][r];
            H[(long long)row * OUT + col]   = v;
            agg[(long long)row * OUT + col] = v * d2 + bias[col];
        }
    }
}

// ---------------- edge scatter ----------------
// agg[dst] += h[src] * norm[e]; OUT/4 threads per edge, float4 gathers.
template <int OUT>
__global__ void k_scatter(const int* __restrict__ src, const int* __restrict__ dst,
                          const float* __restrict__ norm, const float* __restrict__ H,
                          float* __restrict__ agg) {
    constexpr int TPE = OUT / 4;
    int gid = blockIdx.x * blockDim.x + threadIdx.x;
    if (gid >= NE * TPE) return;
    const int e = gid / TPE;
    const int c = (gid % TPE) * 4;
    const float w = norm[e];
    const float4 h4 = *(const float4*)(H + (long long)src[e] * OUT + c);
    float* p = agg + (long long)dst[e] * OUT + c;
    atomic_add_f32(p + 0, h4.x * w);
    atomic_add_f32(p + 1, h4.y * w);
    atomic_add_f32(p + 2, h4.z * w);
    atomic_add_f32(p + 3, h4.w * w);
}

// ---------------- epilogues ----------------
// SELU only (bias already folded into agg by the GEMM epilogue).
__global__ void k_selu(const float* __restrict__ agg, float* __restrict__ out, int total) {
    int gid = blockIdx.x * blockDim.x + threadIdx.x;
    if (gid < total) {
        const float v = agg[gid];
        const float alpha = 1.6732632423543772f;
        const float scale = 1.0507009873554805f;
        out[gid] = scale * (v > 0.0f ? v : alpha * (expf(v) - 1.0f));
    }
}

// Last layer: L2 row-normalize (rows of 64), one wave per row (wave32).
__global__ __launch_bounds__(256) void k_l2norm(const float* __restrict__ agg,
                                                float* __restrict__ out) {
    const int row  = blockIdx.x * 8 + (threadIdx.x >> 5);
    const int lane = threadIdx.x & 31;
    const float v0 = agg[(long long)row * 64 + lane];
    const float v1 = agg[(long long)row * 64 + 32 + lane];
    float s = v0 * v0 + v1 * v1;
#pragma unroll
    for (int off = 16; off >= 1; off >>= 1) s += __shfl_xor(s, off, 32);
    const float inv = 1.0f / fmaxf(sqrtf(s), 1e-12f);
    out[(long long)row * 64 + lane]      = v0 * inv;
    out[(long long)row * 64 + 32 + lane] = v1 * inv;
}

// ---------------- host orchestration ----------------
extern "C" void kernel_launch(void* const* d_in, const int* in_sizes, int n_in,
                              void* d_out, int out_size, void* d_ws, size_t ws_size,
                              hipStream_t stream) {
    const float* x  = (const float*)d_in[0];
    const int*   ei = (const int*)d_in[1];
    const int* srcI = ei;
    const int* dstI = ei + NE;
    const float* W1 = (const float*)d_in[2]; const float* b1 = (const float*)d_in[3];
    const float* W2 = (const float*)d_in[4]; const float* b2 = (const float*)d_in[5];
    const float* W3 = (const float*)d_in[6]; const float* b3 = (const float*)d_in[7];
    const float* W4 = (const float*)d_in[8]; const float* b4 = (const float*)d_in[9];

    float* ws   = (float*)d_ws;
    float* dinv = ws;                           // NN
    float* norm = dinv + NN;                    // NE
    float* bufH = norm + NE;                    // NN*128
    float* bufA = bufH + (size_t)NN * 128;      // NN*128
    float* bufX = bufA + (size_t)NN * 128;      // NN*128

    const int T = 256;
    // normalization coefficients
    k_deg_init<<<(NN + T - 1) / T, T, 0, stream>>>(dinv);
    k_deg_accum<<<(NE + T - 1) / T, T, 0, stream>>>(dstI, dinv);
    k_dinv<<<(NN + T - 1) / T, T, 0, stream>>>(dinv);
    k_norm<<<(NE + T - 1) / T, T, 0, stream>>>(srcI, dstI, dinv, norm);

    const int MT = NN / 16;  // 3125 row tiles, exact

    // Layer 1: 128 -> 128
    k_gemm_wmma<128, 128><<<MT, 32, 0, stream>>>(x, W1, b1, dinv, bufH, bufA);
    k_scatter<128><<<(NE * 32 + T - 1) / T, T, 0, stream>>>(srcI, dstI, norm, bufH, bufA);
    k_selu<<<(NN * 128 + T - 1) / T, T, 0, stream>>>(bufA, bufX, NN * 128);

    // Layer 2: 128 -> 128
    k_gemm_wmma<128, 128><<<MT, 32, 0, stream>>>(bufX, W2, b2, dinv, bufH, bufA);
    k_scatter<128><<<(NE * 32 + T - 1) / T, T, 0, stream>>>(srcI, dstI, norm, bufH, bufA);
    k_selu<<<(NN * 128 + T - 1) / T, T, 0, stream>>>(bufA, bufX, NN * 128);

    // Layer 3: 128 -> 64
    k_gemm_wmma<128, 64><<<MT, 32, 0, stream>>>(bufX, W3, b3, dinv, bufH, bufA);
    k_scatter<64><<<(NE * 16 + T - 1) / T, T, 0, stream>>>(srcI, dstI, norm, bufH, bufA);
    k_selu<<<(NN * 64 + T - 1) / T, T, 0, stream>>>(bufA, bufX, NN * 64);

    // Layer 4: 64 -> 64, then L2 normalize into d_out (bias already in agg)
    k_gemm_wmma<64, 64><<<MT, 32, 0, stream>>>(bufX, W4, b4, dinv, bufH, bufA);
    k_scatter<64><<<(NE * 16 + T - 1) / T, T, 0, stream>>>(srcI, dstI, norm, bufH, bufA);
    k_l2norm<<<NN / 8, 256, 0, stream>>>(bufA, (float*)d_out);
}